// SpectralSelfAttention_21268678049974
// MI455X (gfx1250) — compile-verified
//
#include <hip/hip_runtime.h>
#include <hip/hip_bf16.h>
#include <math.h>
#include <stdint.h>

// ---------------------------------------------------------------------------
// SpectralSelfAttention for MI455X (gfx1250), fp32 end-to-end.
// - All GEMM-shaped stages use V_WMMA_F32_16X16X4_F32 (wave32 fragments).
// - LDS tiles staged with GLOBAL_LOAD_ASYNC_TO_LDS_B128 (ASYNCcnt),
//   double-buffered so HBM fetch overlaps WMMA compute.
// ---------------------------------------------------------------------------

typedef __attribute__((ext_vector_type(2))) float v2f;
typedef __attribute__((ext_vector_type(8))) float v8f;

#define DIM     384
#define HEADS   8
#define CH      48            // DIM / HEADS
#define BATCH   4
#define NPIX    16384         // 128*128
#define C3      1152          // 3*DIM
#define CLAMP_V 50.0f
#define EPSF    1e-12f

// ---- CDNA5 async copy helpers ---------------------------------------------
// Per-lane: copies 16 bytes from the global address gptr to the LDS address lds_off (tracked with ASYNCcnt)
__device__ __forceinline__ void async_ld_b128(unsigned lds_off, const void* gptr) {
    asm volatile("global_load_async_to_lds_b128 %0, %1, off"
                 :: "v"(lds_off), "v"(gptr) : "memory");
}
__device__ __forceinline__ void wait_async0() {
    asm volatile("s_wait_asynccnt 0" ::: "memory");
}
// The low 32 bits of a flat LDS pointer are the LDS byte offset; the high half is the aperture
__device__ __forceinline__ unsigned lds_off(const void* p) {
    return (unsigned)(uintptr_t)p;
}

// ---------------------------------------------------------------------------
// Kernel 1: spectral-norm scale per weight: scale = 1 / (sigma + eps),
// sigma = || W * unit(W^T u) ||   (one power iteration, matching reference).
// blockIdx.x: 0 = qkv_w (1152x384), 1 = dw_w (1152x9), 2 = proj_w (384x384)
// ---------------------------------------------------------------------------
__global__ __launch_bounds__(256)
void sigma_kernel(const float* __restrict__ qkv_w, const float* __restrict__ dw_w,
                  const float* __restrict__ proj_w,
                  const float* __restrict__ u_qkv, const float* __restrict__ u_dw,
                  const float* __restrict__ u_proj, float* __restrict__ scales) {
    __shared__ float sv[DIM];
    __shared__ float red[256];
    const float* W; const float* u; int rows, cols;
    if (blockIdx.x == 0)      { W = qkv_w;  u = u_qkv;  rows = C3;  cols = DIM; }
    else if (blockIdx.x == 1) { W = dw_w;   u = u_dw;   rows = C3;  cols = 9;   }
    else                      { W = proj_w; u = u_proj; rows = DIM; cols = DIM; }
    int t = threadIdx.x;

    for (int j = t; j < cols; j += 256) {            // t_j = sum_i W[i,j] u[i]
        float acc = 0.f;
        for (int i = 0; i < rows; ++i) acc += W[i * cols + j] * u[i];
        sv[j] = acc;
    }
    __syncthreads();
    float p = 0.f;
    for (int j = t; j < cols; j += 256) p += sv[j] * sv[j];
    red[t] = p; __syncthreads();
    for (int s = 128; s > 0; s >>= 1) { if (t < s) red[t] += red[t + s]; __syncthreads(); }
    float inv = 1.f / fmaxf(sqrtf(red[0]), EPSF);
    __syncthreads();
    for (int j = t; j < cols; j += 256) sv[j] *= inv;   // v = unit(W^T u)
    __syncthreads();

    float acc2 = 0.f;                                   // sigma^2 = sum_i (W v)_i^2
    for (int i = t; i < rows; i += 256) {
        float si = 0.f;
        for (int j = 0; j < cols; ++j) si += W[i * cols + j] * sv[j];
        acc2 += si * si;
    }
    red[t] = acc2; __syncthreads();
    for (int s = 128; s > 0; s >>= 1) { if (t < s) red[t] += red[t + s]; __syncthreads(); }
    if (t == 0) scales[blockIdx.x] = 1.f / (sqrtf(red[0]) + EPSF);
}

// ---------------------------------------------------------------------------
// Kernel 2/7: 1x1 conv as fp32-WMMA GEMM with async double-buffered staging.
// Out[b, m, n] = scale * sum_k Wt[m, k] * X[b, k, n]
// Block = 256 threads (8 waves); block tile = 128(M) x 64(N); K chunked by 16.
// LDS row strides: W=20 floats (80B, 16B-aligned, conflict-free A-frag),
//                  X=80 floats (320B, conflict-free B-frag halves).
// ---------------------------------------------------------------------------
template <int M, int K>
__global__ __launch_bounds__(256)
void conv1x1_wmma(const float* __restrict__ Wt, const float* __restrict__ X,
                  float* __restrict__ Out, const float* __restrict__ scales, int sidx) {
    __shared__ float Ws[2][128][20];
    __shared__ float Xs[2][16][80];
    const int n0 = blockIdx.x * 64;
    const int m0 = blockIdx.y * 128;
    const int b  = blockIdx.z;
    const int t = threadIdx.x, lane = t & 31, w = t >> 5;
    const float* Xb = X + (size_t)b * K * NPIX;

    v8f acc[4] = {};
    const int l15  = lane & 15;
    const int ksel = (lane >> 4) * 2;          // 0 for lanes 0-15, 2 for 16-31

    auto stage = [&](int buf, int k0) {
        // W tile: 128 rows x 16 floats = 512 x b128 chunks, 2 per thread
        #pragma unroll
        for (int it = 0; it < 2; ++it) {
            int idx = t + it * 256;
            int r = idx >> 2, c4 = idx & 3;
            async_ld_b128(lds_off(&Ws[buf][r][c4 * 4]),
                          &Wt[(size_t)(m0 + r) * K + k0 + c4 * 4]);
        }
        // X tile: 16 rows x 64 floats = 256 x b128 chunks, 1 per thread
        {
            int kk = t >> 4, c4 = t & 15;
            async_ld_b128(lds_off(&Xs[buf][kk][c4 * 4]),
                          &Xb[(size_t)(k0 + kk) * NPIX + n0 + c4 * 4]);
        }
    };

    constexpr int NCHUNK = K / 16;
    stage(0, 0);
    wait_async0();
    __syncthreads();

    for (int c = 0; c < NCHUNK; ++c) {
        const int cur = c & 1;
        if (c + 1 < NCHUNK) stage(cur ^ 1, (c + 1) * 16);   // prefetch next chunk

        const int arow = w * 16 + l15;
#pragma unroll
        for (int kk = 0; kk < 16; kk += 4) {
            v2f a;
            a.x = Ws[cur][arow][kk + ksel];
            a.y = Ws[cur][arow][kk + ksel + 1];
#pragma unroll
            for (int j = 0; j < 4; ++j) {
                v2f bf;
                bf.x = Xs[cur][kk + ksel][j * 16 + l15];
                bf.y = Xs[cur][kk + ksel + 1][j * 16 + l15];
                acc[j] = __builtin_amdgcn_wmma_f32_16x16x4_f32(
                    false, a, false, bf, (short)0, acc[j], false, false);
            }
        }
        wait_async0();      // prefetch landed during compute
        __syncthreads();    // everyone done with `cur` before it is re-staged
    }

    const float s = scales[sidx];
    const int mrow = m0 + w * 16 + ((lane >> 4) << 3);   // C: M=r (+8 upper half)
    float* Ob = Out + (size_t)b * M * NPIX;
#pragma unroll
    for (int j = 0; j < 4; ++j)
#pragma unroll
        for (int r = 0; r < 8; ++r)
            Ob[(size_t)(mrow + r) * NPIX + n0 + j * 16 + l15] = acc[j][r] * s;
}

// ---------------------------------------------------------------------------
// Kernel 3: 3x3 depthwise conv, padding=1 (SAME). One thread per pixel,
// channel on grid.y so the 9 weights are wave-uniform (scalarized).
// ---------------------------------------------------------------------------
__global__ __launch_bounds__(256)
void dwconv_kernel(const float* __restrict__ src, const float* __restrict__ dw_w,
                   const float* __restrict__ scales, float* __restrict__ dst) {
    const int c = blockIdx.y;
    const int b = blockIdx.z;
    const int pix = blockIdx.x * 256 + threadIdx.x;
    const int y = pix >> 7, x = pix & 127;
    const float s = scales[1];
    const float* wp = dw_w + (size_t)c * 9;
    const float* sp = src + ((size_t)b * C3 + c) * NPIX;
    float acc = 0.f;
#pragma unroll
    for (int dy = -1; dy <= 1; ++dy) {
        int yy = y + dy;
        if (yy < 0 || yy > 127) continue;
#pragma unroll
        for (int dx = -1; dx <= 1; ++dx) {
            int xx = x + dx;
            if (xx < 0 || xx > 127) continue;
            acc += wp[(dy + 1) * 3 + (dx + 1)] * sp[yy * 128 + xx];
        }
    }
    dst[((size_t)b * C3 + c) * NPIX + pix] = acc * s;
}

// ---------------------------------------------------------------------------
// Kernel 4: reciprocal L2 norms of q and k rows (length NPIX).
// ---------------------------------------------------------------------------
__global__ __launch_bounds__(256)
void rnorm_kernel(const float* __restrict__ qkv, float* __restrict__ rq,
                  float* __restrict__ rk) {
    __shared__ float red[256];
    const int id = blockIdx.x;
    const int which = id / (BATCH * DIM);      // 0=q, 1=k
    const int rem = id % (BATCH * DIM);
    const int b = rem / DIM, c = rem % DIM;
    const float* row = qkv + ((size_t)b * C3 + which * DIM + c) * NPIX;
    const int t = threadIdx.x;
    float p = 0.f;
    for (int i = t; i < NPIX; i += 256) { float v = row[i]; p += v * v; }
    red[t] = p; __syncthreads();
    for (int s = 128; s > 0; s >>= 1) { if (t < s) red[t] += red[t + s]; __syncthreads(); }
    if (t == 0) {
        float inv = 1.f / fmaxf(sqrtf(red[0]), EPSF);
        (which ? rk : rq)[b * DIM + c] = inv;
    }
}

// ---------------------------------------------------------------------------
// Kernel 5: attn[b,h,i,j] = softmax_j( clip( (q_i . k_j) rq_i rk_j temp_h ) )
// One block per (b,h). 9 waves: wave w owns 16x16 tile (i0=16*(w/3), j0=16*(w%3)).
// K = NPIX reduced in 64-wide chunks, async double-buffered into LDS.
// ---------------------------------------------------------------------------
__global__ __launch_bounds__(288)
void attn_kernel(const float* __restrict__ qkv, const float* __restrict__ rq,
                 const float* __restrict__ rk, const float* __restrict__ temperature,
                 float* __restrict__ attn) {
    __shared__ float Qs[2][48][68];     // 272B rows: 16B-aligned, conflict-free
    __shared__ float Ks[2][48][68];
    __shared__ float As[48][52];
    const int bh = blockIdx.x, b = bh >> 3, h = bh & 7;
    const int t = threadIdx.x, lane = t & 31, w = t >> 5;   // 9 waves
    const int l15 = lane & 15, ksel = (lane >> 4) * 2;
    const int i0 = (w / 3) * 16, j0 = (w % 3) * 16;
    const float* Qrow = qkv + ((size_t)b * C3 + h * CH) * NPIX;
    const float* Krow = qkv + ((size_t)b * C3 + DIM + h * CH) * NPIX;

    auto stage = [&](int buf, int n0) {
        // 48 rows x 64 floats = 768 b128 chunks per matrix
        for (int idx = t; idx < 768; idx += 288) {
            int r = idx >> 4, c4 = idx & 15;
            async_ld_b128(lds_off(&Qs[buf][r][c4 * 4]),
                          &Qrow[(size_t)r * NPIX + n0 + c4 * 4]);
            async_ld_b128(lds_off(&Ks[buf][r][c4 * 4]),
                          &Krow[(size_t)r * NPIX + n0 + c4 * 4]);
        }
    };

    v8f acc = {};
    stage(0, 0);
    wait_async0();
    __syncthreads();

    for (int c = 0; c < NPIX / 64; ++c) {
        const int cur = c & 1;
        if (c + 1 < NPIX / 64) stage(cur ^ 1, (c + 1) * 64);
#pragma unroll
        for (int kk = 0; kk < 64; kk += 4) {
            v2f a, bf;
            a.x  = Qs[cur][i0 + l15][kk + ksel];
            a.y  = Qs[cur][i0 + l15][kk + ksel + 1];
            bf.x = Ks[cur][j0 + l15][kk + ksel];      // B[n,j] = K[j,n]
            bf.y = Ks[cur][j0 + l15][kk + ksel + 1];
            acc = __builtin_amdgcn_wmma_f32_16x16x4_f32(
                false, a, false, bf, (short)0, acc, false, false);
        }
        wait_async0();
        __syncthreads();
    }

    // scale, clamp, stash in LDS
    const float tmp = temperature[h];
    const int ii = i0 + ((lane >> 4) << 3);
    const int jj = j0 + l15;
    const float rkj = rk[b * DIM + h * CH + jj];
#pragma unroll
    for (int r = 0; r < 8; ++r) {
        float v = acc[r] * rq[b * DIM + h * CH + ii + r] * rkj * tmp;
        As[ii + r][jj] = fminf(fmaxf(v, -CLAMP_V), CLAMP_V);
    }
    __syncthreads();

    // row softmax (48 rows of 48)
    if (t < 48) {
        float m = -1e30f;
        for (int j = 0; j < 48; ++j) m = fmaxf(m, As[t][j]);
        float ssum = 0.f;
        for (int j = 0; j < 48; ++j) { float e = expf(As[t][j] - m); As[t][j] = e; ssum += e; }
        float inv = 1.f / ssum;
        for (int j = 0; j < 48; ++j) As[t][j] *= inv;
    }
    __syncthreads();
    float* ab = attn + (size_t)bh * 48 * 48;
    for (int idx = t; idx < 48 * 48; idx += 288) ab[idx] = As[idx / 48][idx % 48];
}

// ---------------------------------------------------------------------------
// Kernel 6: out_pre[b, h*48+i, n] = sum_j attn[b,h,i,j] * v[b,h,j,n]
// Block: (n-tile of 64) x (b,h); 12 waves, each a 16x16 tile; K = 48.
// Both LDS tiles staged with async B128.
// ---------------------------------------------------------------------------
__global__ __launch_bounds__(384)
void av_kernel(const float* __restrict__ qkv, const float* __restrict__ attn,
               float* __restrict__ outp) {
    __shared__ float As[48][52];
    __shared__ float Vs[48][80];
    const int n0 = blockIdx.x * 64;
    const int bh = blockIdx.y, b = bh >> 3, h = bh & 7;
    const int t = threadIdx.x, lane = t & 31, w = t >> 5;   // 12 waves
    const int l15 = lane & 15, ksel = (lane >> 4) * 2;
    const float* ab = attn + (size_t)bh * 2304;
    const float* Vrow = qkv + ((size_t)b * C3 + 2 * DIM + h * CH) * NPIX;

    // attn: 48 rows x 48 floats = 576 b128 chunks
    for (int idx = t; idx < 576; idx += 384) {
        int r = idx / 12, c4 = idx % 12;
        async_ld_b128(lds_off(&As[r][c4 * 4]), &ab[(size_t)r * 48 + c4 * 4]);
    }
    // V tile: 48 rows x 64 floats = 768 b128 chunks, 2 per thread
#pragma unroll
    for (int it = 0; it < 2; ++it) {
        int idx = t + it * 384;
        int r = idx >> 4, c4 = idx & 15;
        async_ld_b128(lds_off(&Vs[r][c4 * 4]),
                      &Vrow[(size_t)r * NPIX + n0 + c4 * 4]);
    }
    wait_async0();
    __syncthreads();

    const int i0 = (w % 3) * 16, nt = (w / 3) * 16;
    v8f acc = {};
#pragma unroll
    for (int kk = 0; kk < 48; kk += 4) {
        v2f a, bf;
        a.x  = As[i0 + l15][kk + ksel];
        a.y  = As[i0 + l15][kk + ksel + 1];
        bf.x = Vs[kk + ksel][nt + l15];
        bf.y = Vs[kk + ksel + 1][nt + l15];
        acc = __builtin_amdgcn_wmma_f32_16x16x4_f32(
            false, a, false, bf, (short)0, acc, false, false);
    }

    const int ii = i0 + ((lane >> 4) << 3);
    float* ob = outp + ((size_t)b * DIM + h * CH) * NPIX;
#pragma unroll
    for (int r = 0; r < 8; ++r)
        ob[(size_t)(ii + r) * NPIX + n0 + nt + l15] = acc[r];
}

// ---------------------------------------------------------------------------
// Host-side launch. Workspace layout (floats):
//   [0..3)        : scales (qkv, dw, proj)
//   [16..1552)    : rq   [1552..3088) : rk   [3088..76816) : attn
//   [81920 ..)    : bufA = B*C3*NPIX   (qkv_pre; later reused as attn-out)
//   bufA + 75497472 : bufB = B*C3*NPIX (qkv after depthwise)   (~605 MB total)
// ---------------------------------------------------------------------------
extern "C" void kernel_launch(void* const* d_in, const int* in_sizes, int n_in,
                              void* d_out, int out_size, void* d_ws, size_t ws_size,
                              hipStream_t stream) {
    const float* x      = (const float*)d_in[0];
    const float* qkv_w  = (const float*)d_in[1];
    const float* dw_w   = (const float*)d_in[2];
    const float* proj_w = (const float*)d_in[3];
    const float* temp   = (const float*)d_in[4];
    const float* u_qkv  = (const float*)d_in[5];
    const float* u_dw   = (const float*)d_in[6];
    const float* u_proj = (const float*)d_in[7];

    float* ws     = (float*)d_ws;
    float* scales = ws;
    float* rq     = ws + 16;
    float* rk     = ws + 16 + 1536;
    float* attn   = ws + 16 + 3072;
    float* bufA   = ws + 81920;
    float* bufB   = bufA + (size_t)BATCH * C3 * NPIX;

    sigma_kernel<<<3, 256, 0, stream>>>(qkv_w, dw_w, proj_w, u_qkv, u_dw, u_proj, scales);

    conv1x1_wmma<C3, DIM><<<dim3(NPIX / 64, C3 / 128, BATCH), 256, 0, stream>>>(
        qkv_w, x, bufA, scales, 0);

    dwconv_kernel<<<dim3(NPIX / 256, C3, BATCH), 256, 0, stream>>>(bufA, dw_w, scales, bufB);

    rnorm_kernel<<<2 * BATCH * DIM, 256, 0, stream>>>(bufB, rq, rk);

    attn_kernel<<<BATCH * HEADS, 288, 0, stream>>>(bufB, rq, rk, temp, attn);

    av_kernel<<<dim3(NPIX / 64, BATCH * HEADS), 384, 0, stream>>>(bufB, attn, bufA);

    conv1x1_wmma<DIM, DIM><<<dim3(NPIX / 64, DIM / 128, BATCH), 256, 0, stream>>>(
        proj_w, bufA, (float*)d_out, scales, 2);
}